// BeliefPropagationSingle_61564061221585
// MI455X (gfx1250) — compile-verified
//
#include <hip/hip_runtime.h>

#define NMAT 16
#define DDIM 2048

typedef float v2f __attribute__((ext_vector_type(2)));
typedef float v8f __attribute__((ext_vector_type(8)));

// ---------------------------------------------------------------------------
// Zero rowsums each call (ws is poisoned; atomics below must start from 0).
// ---------------------------------------------------------------------------
__global__ void __launch_bounds__(1024) zero_rowsums_kernel(float* __restrict__ rowsums) {
  rowsums[blockIdx.x * 1024 + threadIdx.x] = 0.f;
}

// ---------------------------------------------------------------------------
// Fused sums kernel: one streaming read of theta serves BOTH reductions.
// Phase 1 — column sums via V_WMMA_F32_16X16X4_F32:
//   D16x16 = ones(16x4) * B(4x16 tile of theta) + C, accumulated over rows.
//   Each wave owns this block's 16-column strip over a 256-row band; the 8
//   waves combine partials in LDS. With A==ones any K-permutation of B is
//   still the sum and every row of D replicates the colsum -> layout-robust.
// Phase 2 — row-sum partials for the SAME 16-col strip (cache-hot re-read):
//   each thread sums 16 cols of 8 rows and global_atomic_add_f32's into
//   rowsums[j][row]. Worst case (cache miss) this equals the old separate
//   kernel's traffic; best case it rides L0/L2 and halves the sums-phase HBM
//   traffic (256 MB saved ~ 11 us at 23.3 TB/s).
// ---------------------------------------------------------------------------
__global__ void __launch_bounds__(256) sums_fused_kernel(
    const float* __restrict__ theta, float* __restrict__ colsums,
    float* __restrict__ rowsums) {
  const int j = blockIdx.y;
  const int colbase = blockIdx.x * 16;
  const int lane = threadIdx.x & 31;
  const int wave = threadIdx.x >> 5;           // 0..7
  const float* __restrict__ T = theta + (size_t)j * DDIM * DDIM;

  // ---- Phase 1: WMMA colsum over this strip ----
  const int colOff = colbase + (lane & 15);
  const int rowSel = (lane >> 4) << 1;         // 0 or 2
  v2f ones2; ones2.x = 1.0f; ones2.y = 1.0f;
  v8f acc = {};
  const int row0 = wave * (DDIM / 8);
#pragma unroll 4
  for (int k = 0; k < DDIM / 8; k += 4) {
    const size_t r = (size_t)(row0 + k + rowSel);
    v2f b;
    b.x = T[r * DDIM + colOff];
    b.y = T[(r + 1) * DDIM + colOff];
    acc = __builtin_amdgcn_wmma_f32_16x16x4_f32(false, ones2, false, b,
                                                (short)0, acc, false, false);
  }
  __shared__ float parts[8][16];
  if (lane < 16) parts[wave][lane] = acc[0];   // row M=0, N=lane of D
  __syncthreads();
  if (threadIdx.x < 16) {
    float s = 0.f;
#pragma unroll
    for (int w = 0; w < 8; ++w) s += parts[w][threadIdx.x];
    colsums[j * DDIM + colbase + threadIdx.x] = s;
  }

  // ---- Phase 2: row-sum partials for this strip (L0/L2-hot) ----
#pragma unroll
  for (int rr = 0; rr < 8; ++rr) {
    const int row = threadIdx.x + 256 * rr;
    const float4* __restrict__ p =
        (const float4*)(T + (size_t)row * DDIM + colbase);
    float s = 0.f;
#pragma unroll
    for (int q = 0; q < 4; ++q) {
      float4 v = p[q];
      s += (v.x + v.y) + (v.z + v.w);
    }
    atomicAdd(&rowsums[j * DDIM + row], s);
  }
}

// ---------------------------------------------------------------------------
// Sequential message recurrences (15 fwd + 15 bwd steps) in ONE workgroup.
// Per step: gather from a [D] vector, scatter-add into LDS (ds_add_f32
// atomics), store message, reduce to the scalar carry for the next step.
//   fwd:  g[e] = colsums[i][ia_f[e]] + S ;  proc_fwd[i] = scatter_add(ib_f, g)
//   bwd:  m[e] = rowsums[i+1][ia_b[e]] + (D-1)*proc_fwd[i][ia_b[e]] + T
// ---------------------------------------------------------------------------
__global__ void __launch_bounds__(1024) messages_kernel(
    const float* __restrict__ colsums, const float* __restrict__ rowsums,
    const int* __restrict__ ia_f, const int* __restrict__ ib_f,
    const int* __restrict__ ia_b, const int* __restrict__ ib_b,
    float* __restrict__ proc_fwd, float* __restrict__ proc_bwd) {
  __shared__ float scat[DDIM];
  __shared__ float sSum;
  const int t = threadIdx.x;
  const int lane = t & 31;

  float carry = 0.f;
  for (int i = 0; i < NMAT - 1; ++i) {          // forward sweep
    if (t == 0) sSum = 0.f;
    scat[t] = 0.f; scat[t + 1024] = 0.f;
    __syncthreads();
    for (int e = t; e < DDIM; e += 1024) {
      const int a = ia_f[i * DDIM + e];
      const float g = colsums[i * DDIM + a] + carry;
      atomicAdd(&scat[ib_f[i * DDIM + e]], g);
    }
    __syncthreads();
    float loc = 0.f;
    for (int e = t; e < DDIM; e += 1024) {
      const float v = scat[e];
      proc_fwd[i * DDIM + e] = v;
      loc += v;
    }
#pragma unroll
    for (int off = 16; off > 0; off >>= 1) loc += __shfl_xor(loc, off);
    if (lane == 0) atomicAdd(&sSum, loc);
    __syncthreads();
    carry = sSum;
    __syncthreads();
  }

  carry = 0.f;
  for (int i = NMAT - 2; i >= 0; --i) {         // backward sweep
    if (t == 0) sSum = 0.f;
    scat[t] = 0.f; scat[t + 1024] = 0.f;
    __syncthreads();
    for (int e = t; e < DDIM; e += 1024) {
      const int a = ia_b[i * DDIM + e];
      const float m = rowsums[(i + 1) * DDIM + a]
                    + (float)(DDIM - 1) * proc_fwd[i * DDIM + a]
                    + carry;
      atomicAdd(&scat[ib_b[i * DDIM + e]], m);
    }
    __syncthreads();
    float loc = 0.f;
    for (int e = t; e < DDIM; e += 1024) {
      const float v = scat[e];
      proc_bwd[i * DDIM + e] = v;
      loc += v;
    }
#pragma unroll
    for (int off = 16; off > 0; off >>= 1) loc += __shfl_xor(loc, off);
    if (lane == 0) atomicAdd(&sSum, loc);
    __syncthreads();
    carry = sSum;
    __syncthreads();
  }
}

// ---------------------------------------------------------------------------
// Streaming output: out[j] = theta[j] + proc_fwd[j-1][row] + proc_bwd[j][col].
// float4 vectorized read/modify/write; messages stay resident in cache.
// ---------------------------------------------------------------------------
__global__ void __launch_bounds__(256) apply_kernel(
    const float* __restrict__ theta, const float* __restrict__ proc_fwd,
    const float* __restrict__ proc_bwd, float* __restrict__ out) {
  const int gid = blockIdx.x * 256 + threadIdx.x;      // float4 units
  const int perMat = (DDIM * DDIM) / 4;
  const int j = gid / perMat;
  const int rem = gid - j * perMat;
  const int row = rem / (DDIM / 4);
  const int c4 = rem - row * (DDIM / 4);

  float4 v = ((const float4*)theta)[gid];
  const float pf = (j >= 1) ? proc_fwd[(j - 1) * DDIM + row] : 0.f;
  float4 pb = make_float4(0.f, 0.f, 0.f, 0.f);
  if (j <= NMAT - 2) pb = ((const float4*)proc_bwd)[j * (DDIM / 4) + c4];
  v.x += pf + pb.x; v.y += pf + pb.y; v.z += pf + pb.z; v.w += pf + pb.w;
  ((float4*)out)[gid] = v;
}

extern "C" void kernel_launch(void* const* d_in, const int* in_sizes, int n_in,
                              void* d_out, int out_size, void* d_ws, size_t ws_size,
                              hipStream_t stream) {
  const float* theta = (const float*)d_in[0];
  const int* ia_f = (const int*)d_in[1];
  const int* ib_f = (const int*)d_in[2];
  const int* ia_b = (const int*)d_in[3];
  const int* ib_b = (const int*)d_in[4];
  float* out = (float*)d_out;

  float* ws = (float*)d_ws;
  float* colsums  = ws;                             // N*D floats
  float* rowsums  = colsums + NMAT * DDIM;          // N*D floats
  float* proc_fwd = rowsums + NMAT * DDIM;          // (N-1)*D floats
  float* proc_bwd = proc_fwd + (NMAT - 1) * DDIM;   // (N-1)*D floats

  zero_rowsums_kernel<<<(NMAT * DDIM) / 1024, 1024, 0, stream>>>(rowsums);
  sums_fused_kernel<<<dim3(DDIM / 16, NMAT), 256, 0, stream>>>(theta, colsums,
                                                               rowsums);
  messages_kernel<<<1, 1024, 0, stream>>>(colsums, rowsums, ia_f, ib_f, ia_b,
                                          ib_b, proc_fwd, proc_bwd);
  const int n4 = NMAT * DDIM * DDIM / 4;
  apply_kernel<<<n4 / 256, 256, 0, stream>>>(theta, proc_fwd, proc_bwd, out);
}